// WaveGainLayer_20882130993338
// MI455X (gfx1250) — compile-verified
//
#include <hip/hip_runtime.h>

typedef __attribute__((ext_vector_type(2))) float v2f;
typedef __attribute__((ext_vector_type(8))) float v8f;

// D = A(16x4) * B(4x16) + C, fp32, wave32 WMMA.
__device__ __forceinline__ v8f wmma_f32(v2f a, v2f b, v8f c) {
  return __builtin_amdgcn_wmma_f32_16x16x4_f32(
      /*neg_a=*/false, a, /*neg_b=*/false, b,
      /*c_mod=*/(short)0, c, /*reuse_a=*/false, /*reuse_b=*/false);
}

// Async global->LDS b64 copy (GV mode, per-lane 64-bit address), ASYNCcnt.
// LDS destination is the low 32 bits of the generic pointer (LDS aperture
// truncation, ISA 10.2). Completion: s_wait_asynccnt 0 + barrier.
__device__ __forceinline__ void async_copy_b64(float* lds_dst,
                                               const float* gsrc) {
  const unsigned lds_off = (unsigned)(size_t)lds_dst;
  asm volatile("global_load_async_to_lds_b64 %0, %1, off"
               :
               : "v"(lds_off), "v"(gsrc)
               : "memory");
}

__device__ __forceinline__ void wait_async0() {
  asm volatile("s_wait_asynccnt 0x0" ::: "memory");
}

// ---------------------------------------------------------------------------
// Band complex-gain kernel: 1x1 complex channel mix as a real GEMM, K' = 2C.
//   u: (N, C, 6, H, H, 2)   g: (6, 2, F, C)   v: (N, F, 6, H, H, 2)
// Workgroup: 256 threads (8 waves). Tile: all F=128 x 64 pixels (u read once).
// Wave w computes f in [16w, 16w+16), all 64 pixels (4 N-subtiles), with
// paired accumulators (vr, vi) sharing every B fragment. Both A fragments
// (A'_r = {gr,-gi}, A'_i = {gi,gr}) are built on the fly from one {gr,gi}
// LDS pair, so only one gain array lives in LDS (50.7 KB/WG -> 6 WG/WGP).
// ---------------------------------------------------------------------------
template <int H>
__global__ __launch_bounds__(256) void band_gain_wmma(
    const float* __restrict__ u, const float* __restrict__ g,
    float* __restrict__ v) {
  constexpr int C = 128, F = 128;
  constexpr int PX = H * H;
  constexpr int KCH = 32;        // real channels per K-chunk
  constexpr int KC = 2 * KCH;    // K' elements per chunk (=64)
  constexpr int PITCH = KC + 2;  // LDS row pitch (floats), even for b64

  __shared__ float sG[F * PITCH];    // [f][2c..2c+1] = {gr, gi}
  __shared__ float sBT[64 * PITCH];  // B'^T[px][k'] = {ur, ui} interleaved

  const int pxTile = blockIdx.x;  // PX/64 tiles
  const int nl = blockIdx.y;      // n*6 + l
  const int n = nl / 6, l = nl % 6;
  const int pxBase = pxTile * 64;

  const int tid = threadIdx.x;
  const int lane = tid & 31;
  const int wave = tid >> 5;  // 0..7 -> f subtile
  const int lm = lane & 15;
  const int kh = lane >> 4;   // K half select (K{0,1} vs K{2,3})

  v8f accr[4] = {v8f{}, v8f{}, v8f{}, v8f{}};
  v8f acci[4] = {v8f{}, v8f{}, v8f{}, v8f{}};

  // u addressing: ((((n*C + c)*6 + l)*PX + px)*2 + ri)
  const size_t uCS = (size_t)6 * PX * 2;  // channel stride
  const float* uBase = u + (((size_t)n * C * 6 + l) * PX + pxBase) * 2;
  const float* grBase = g + (size_t)(l * 2 + 0) * F * C;
  const float* giBase = g + (size_t)(l * 2 + 1) * F * C;

  for (int cb = 0; cb < C; cb += KCH) {
    __syncthreads();  // prior-chunk LDS reads done before new async writes

    // ---- async stage B'^T: 64 px x 32 channels, contiguous {ur,ui} b64 ----
    {
      const int px = tid & 63;
      const int c0 = tid >> 6;  // 0..3
#pragma unroll
      for (int i = 0; i < 8; ++i) {
        const int c = c0 + 4 * i;  // 0..31
        const float* p = uBase + (size_t)(cb + c) * uCS + (size_t)px * 2;
        async_copy_b64(&sBT[px * PITCH + 2 * c], p);
      }
    }
    // ---- stage gains: 128 f x 32 channels as {gr, gi} pairs ----
    {
      const int c = tid & 31;
      const int f0 = tid >> 5;  // 0..7
#pragma unroll
      for (int i = 0; i < 16; ++i) {
        const int f = f0 + 8 * i;  // 0..127
        const float gr = grBase[(size_t)f * C + cb + c];
        const float gi = giBase[(size_t)f * C + cb + c];
        float* d = &sG[f * PITCH + 2 * c];
        d[0] = gr;
        d[1] = gi;
      }
    }
    wait_async0();
    __syncthreads();

    const float* gRow = &sG[(wave * 16 + lm) * PITCH + 2 * kh];
    const float* bRow0 = &sBT[(0 * 16 + lm) * PITCH + 2 * kh];
    const float* bRow1 = &sBT[(1 * 16 + lm) * PITCH + 2 * kh];
    const float* bRow2 = &sBT[(2 * 16 + lm) * PITCH + 2 * kh];
    const float* bRow3 = &sBT[(3 * 16 + lm) * PITCH + 2 * kh];

#pragma unroll
    for (int ks = 0; ks < KC / 4; ++ks) {
      const v2f gg = *(const v2f*)(gRow + ks * 4);
      v2f ar;  // A'_r fragment: {gr, -gi}  (neg_a would negate gr too)
      ar.x = gg.x;
      ar.y = -gg.y;
      v2f ai;  // A'_i fragment: {gi, gr}
      ai.x = gg.y;
      ai.y = gg.x;
      const v2f b0 = *(const v2f*)(bRow0 + ks * 4);
      const v2f b1 = *(const v2f*)(bRow1 + ks * 4);
      const v2f b2 = *(const v2f*)(bRow2 + ks * 4);
      const v2f b3 = *(const v2f*)(bRow3 + ks * 4);
      accr[0] = wmma_f32(ar, b0, accr[0]);
      acci[0] = wmma_f32(ai, b0, acci[0]);
      accr[1] = wmma_f32(ar, b1, accr[1]);
      acci[1] = wmma_f32(ai, b1, acci[1]);
      accr[2] = wmma_f32(ar, b2, accr[2]);
      acci[2] = wmma_f32(ai, b2, acci[2]);
      accr[3] = wmma_f32(ar, b3, accr[3]);
      acci[3] = wmma_f32(ai, b3, acci[3]);
    }
  }

  // ---- store: lane writes interleaved {vr, vi} float2, coalesced rows ----
  // v index: ((((n*F + f)*6 + l)*PX + px)*2 + ri)
#pragma unroll
  for (int j = 0; j < 4; ++j) {
#pragma unroll
    for (int r = 0; r < 8; ++r) {
      const int f = wave * 16 + r + 8 * kh;  // D row -> output channel
      const int px = pxBase + j * 16 + lm;   // D col -> pixel
      const size_t o = (((size_t)n * F + f) * 6 + l) * PX * 2 + (size_t)px * 2;
      float2 val;
      val.x = accr[j][r];
      val.y = acci[j][r];
      *reinterpret_cast<float2*>(v + o) = val;
    }
  }
}

// ---------------------------------------------------------------------------
// One-shot weight transpose: w (F,C,3,3) -> wT (9, F, C) so LP staging reads
// are coalesced rows instead of stride-9 gathers.
// ---------------------------------------------------------------------------
__global__ __launch_bounds__(256) void transpose_w(
    const float* __restrict__ w, float* __restrict__ wt) {
  constexpr int FC = 128 * 128;
  const int idx = blockIdx.x * 256 + threadIdx.x;  // over 9*F*C outputs
  if (idx < 9 * FC) {
    const int tap = idx / FC;
    const int fc = idx - tap * FC;
    wt[idx] = w[(size_t)fc * 9 + tap];
  }
}

// ---------------------------------------------------------------------------
// Low-pass 3x3 conv as implicit GEMM (tap loop x C-chunks), fp32 WMMA.
//   u: (N, C, 32, 32)   weights: see TRANSPOSED   v: (N, F, 32, 32)
// Workgroup tile: 64 f x 64 px (2 image rows). 8 waves: wave = (wf, wp),
// wf in 0..3 picks 16-f subtile, wp in 0..1 picks 32-px half (2 N-subtiles).
// ---------------------------------------------------------------------------
template <bool TRANSPOSED>
__global__ __launch_bounds__(256) void lp_conv_wmma(
    const float* __restrict__ u, const float* __restrict__ w,
    float* __restrict__ v) {
  constexpr int C = 128, F = 128, W = 32, HI = 32;
  constexpr int KCH = 32;
  constexpr int PITCH = KCH + 4;  // 36, even

  __shared__ float sB[64 * PITCH];  // shifted input tile [px][c]
  __shared__ float sA[64 * PITCH];  // weights for tap     [f][c]

  const int pxTile = blockIdx.x;  // 16 tiles of 2 rows
  const int fBase = blockIdx.y * 64;
  const int n = blockIdx.z;
  const int y0 = pxTile * 2;

  const int tid = threadIdx.x;
  const int lane = tid & 31;
  const int wave = tid >> 5;
  const int wf = wave >> 1;  // 0..3
  const int wp = wave & 1;   // 0..1
  const int lm = lane & 15;
  const int kh = lane >> 4;

  v8f acc0{}, acc1{};

  for (int tap = 0; tap < 9; ++tap) {
    const int ky = tap / 3 - 1;
    const int kx = tap % 3 - 1;
    for (int cb = 0; cb < C; cb += KCH) {
      __syncthreads();
      // ---- stage shifted, zero-padded input tile ----
      {
        const int px = tid & 63;
        const int row = px >> 5, x = px & 31;
        const int y = y0 + row + ky;
        const int xs = x + kx;
        const bool in = (y >= 0) && (y < HI) && (xs >= 0) && (xs < W);
        const int c0 = tid >> 6;
#pragma unroll
        for (int i = 0; i < 8; ++i) {
          const int c = c0 + 4 * i;
          float val = 0.0f;
          if (in) {
            const float* p = &u[(((size_t)n * C + cb + c) * HI + y) * W + xs];
            if (cb + KCH < C) __builtin_prefetch(p + KCH * (HI * W), 0, 1);
            val = *p;
          }
          sB[px * PITCH + c] = val;
        }
      }
      // ---- stage weights for this tap ----
      {
        const int c = tid & 31;
        const int f0 = tid >> 5;
#pragma unroll
        for (int i = 0; i < 8; ++i) {
          const int f = f0 + 8 * i;
          float wv;
          if (TRANSPOSED) {  // wT (9, F, C): coalesced row read
            wv = w[((size_t)tap * F + (fBase + f)) * C + cb + c];
          } else {           // w (F, C, 3, 3): stride-9 gather (L2 resident)
            wv = w[((size_t)(fBase + f) * C + cb + c) * 9 + tap];
          }
          sA[f * PITCH + c] = wv;
        }
      }
      __syncthreads();

      const float* aRow = &sA[(wf * 16 + lm) * PITCH + 2 * kh];
      const float* b0Row = &sB[(wp * 32 + lm) * PITCH + 2 * kh];
      const float* b1Row = &sB[(wp * 32 + 16 + lm) * PITCH + 2 * kh];
#pragma unroll
      for (int ks = 0; ks < KCH / 4; ++ks) {
        const v2f a = *(const v2f*)(aRow + ks * 4);
        const v2f b0 = *(const v2f*)(b0Row + ks * 4);
        const v2f b1 = *(const v2f*)(b1Row + ks * 4);
        acc0 = wmma_f32(a, b0, acc0);
        acc1 = wmma_f32(a, b1, acc1);
      }
    }
  }

  // ---- store: v[((n*F + f)*32 + y)*32 + x] ----
#pragma unroll
  for (int r = 0; r < 8; ++r) {
    const int f = fBase + wf * 16 + r + 8 * kh;
    const int px = y0 * W + wp * 32 + lm;
    v[((size_t)n * F + f) * (HI * W) + px] = acc0[r];
    v[((size_t)n * F + f) * (HI * W) + px + 16] = acc1[r];
  }
}

// ---------------------------------------------------------------------------
extern "C" void kernel_launch(void* const* d_in, const int* in_sizes, int n_in,
                              void* d_out, int out_size, void* d_ws,
                              size_t ws_size, hipStream_t stream) {
  (void)in_sizes; (void)n_in; (void)out_size;

  const float* u_lp = (const float*)d_in[0];  // (8,128,32,32)
  const float* u1   = (const float*)d_in[1];  // (8,128,6,64,64,2)
  const float* u2   = (const float*)d_in[2];  // (8,128,6,32,32,2)
  const float* g_lp = (const float*)d_in[3];  // (128,128,3,3)
  const float* g1   = (const float*)d_in[4];  // (6,2,128,128,1,1)
  const float* g2   = (const float*)d_in[5];  // (6,2,128,128,1,1)

  float* out = (float*)d_out;
  float* v_lp = out;                                   // 8*128*32*32
  float* v1 = v_lp + (size_t)8 * 128 * 32 * 32;        // 8*128*6*64*64*2
  float* v2 = v1 + (size_t)8 * 128 * 6 * 64 * 64 * 2;  // 8*128*6*32*32*2

  const size_t wtBytes = (size_t)9 * 128 * 128 * sizeof(float);
  dim3 blk(256);

  if (ws_size >= wtBytes) {
    float* wT = (float*)d_ws;
    transpose_w<<<dim3((9 * 128 * 128 + 255) / 256), blk, 0, stream>>>(g_lp, wT);
    lp_conv_wmma<true><<<dim3(16, 2, 8), blk, 0, stream>>>(u_lp, wT, v_lp);
  } else {
    lp_conv_wmma<false><<<dim3(16, 2, 8), blk, 0, stream>>>(u_lp, g_lp, v_lp);
  }
  band_gain_wmma<64><<<dim3(64, 8 * 6, 1), blk, 0, stream>>>(u1, g1, v1);
  band_gain_wmma<32><<<dim3(16, 8 * 6, 1), blk, 0, stream>>>(u2, g2, v2);
}